// KoLeoLoss_51290499449142
// MI455X (gfx1250) — compile-verified
//
#include <hip/hip_runtime.h>
#include <stdint.h>

// KoLeo loss, MI455X (gfx1250, wave32).
// Pipeline: row norms -> fp32 WMMA GEMM producing masked d^2 matrix (ws, NT
// stores) -> per-row LDS radix-select of rank-819 (descending) index (NT row
// reads) -> deterministic reduction of 1/(||xi - xj[I] + eps||^2 + 1).
//
// d^2 (256 MiB) is streamed once-write / once-read and does not fit the
// 192 MB L2, while xi/xj (16.8 MB) are reused ~512x and do. Non-temporal
// hints on the d^2 stream keep it from evicting the hot GEMM operands.

typedef float v2f __attribute__((ext_vector_type(2)));
typedef float v8f __attribute__((ext_vector_type(8)));

#define NN 8192
#define DD 256
#define HALF_N (NN / 2)
// descending rank 819  ==  ascending rank NN-1-819
#define RANK_ASC (NN - 1 - (NN / 10))

__device__ __forceinline__ float wave_reduce_sum(float v) {
#pragma unroll
  for (int off = 16; off > 0; off >>= 1) v += __shfl_xor(v, off, 32);
  return v;
}

// ---------------------------------------------------------------- row norms
__global__ __launch_bounds__(256) void koleo_norms(const float* __restrict__ xi,
                                                   const float* __restrict__ xj,
                                                   float* __restrict__ ni,
                                                   float* __restrict__ nj) {
  const int wave = threadIdx.x >> 5, lane = threadIdx.x & 31;
  const int row = blockIdx.x * 8 + wave;  // 0 .. 2*NN-1
  const float* src;
  float* dst;
  int r;
  if (row < NN) { src = xi; dst = ni; r = row; }
  else          { src = xj; dst = nj; r = row - NN; }
  const float* p = src + (size_t)r * DD + lane;
  float s = 0.f;
#pragma unroll
  for (int t = 0; t < DD / 32; ++t) { float v = p[t * 32]; s += v * v; }
  s = wave_reduce_sum(s);
  if (lane == 0) dst[r] = s;
}

// --------------------------------------------- fp32 WMMA GEMM -> masked d^2
// Block: 256 threads = 8 waves. Block tile 16(M) x 512(N); wave tile 16 x 64.
__global__ __launch_bounds__(256) void koleo_gemm_d2(const float* __restrict__ xi,
                                                     const float* __restrict__ xj,
                                                     const float* __restrict__ ni,
                                                     const float* __restrict__ nj,
                                                     float* __restrict__ d2) {
  const int wave  = threadIdx.x >> 5;
  const int lane  = threadIdx.x & 31;
  const int lhalf = lane >> 4;   // 0 or 1
  const int lmod  = lane & 15;
  const int m0 = blockIdx.y * 16;
  const int n0 = blockIdx.x * 512 + wave * 64;

  // A 16x4 f32 fragment: lanes 0-15 carry K=k,k+1; lanes 16-31 carry K=k+2,k+3.
  const float* aPtr = xi + (size_t)(m0 + lmod) * DD + 2 * lhalf;
  // B 4x16 fragment: same K striping, N = lane%16; xj row-major => K contiguous.
  const float* bPtr = xj + (size_t)(n0 + lmod) * DD + 2 * lhalf;

  v8f c0 = {}, c1 = {}, c2 = {}, c3 = {};
#pragma unroll 4
  for (int k = 0; k < DD; k += 4) {
    v2f a  = *(const v2f*)(aPtr + k);
    v2f b0 = *(const v2f*)(bPtr + 0 * 16 * DD + k);
    v2f b1 = *(const v2f*)(bPtr + 1 * 16 * DD + k);
    v2f b2 = *(const v2f*)(bPtr + 2 * 16 * DD + k);
    v2f b3 = *(const v2f*)(bPtr + 3 * 16 * DD + k);
    c0 = __builtin_amdgcn_wmma_f32_16x16x4_f32(false, a, false, b0, (short)0, c0, false, false);
    c1 = __builtin_amdgcn_wmma_f32_16x16x4_f32(false, a, false, b1, (short)0, c1, false, false);
    c2 = __builtin_amdgcn_wmma_f32_16x16x4_f32(false, a, false, b2, (short)0, c2, false, false);
    c3 = __builtin_amdgcn_wmma_f32_16x16x4_f32(false, a, false, b3, (short)0, c3, false, false);
  }

  float njv[4];
#pragma unroll
  for (int t = 0; t < 4; ++t) njv[t] = nj[n0 + t * 16 + lmod];
  v8f cc[4] = {c0, c1, c2, c3};

#pragma unroll
  for (int r = 0; r < 8; ++r) {
    const int row  = m0 + r + 8 * lhalf;     // C/D layout: M = r + 8*(lane/16)
    const float niv = ni[row];
    const int pcol = (row + HALF_N) & (NN - 1);
#pragma unroll
    for (int t = 0; t < 4; ++t) {
      const int col = n0 + t * 16 + lmod;    // N = lane%16
      float v = niv + njv[t] - 2.f * cc[t][r];
      if (col == row || col == pcol) v = -1.f;  // self & positive masks
      // NT store: streamed matrix must not thrash L2 (xi/xj stay resident)
      __builtin_nontemporal_store(v, &d2[(size_t)row * NN + col]);
    }
  }
}

// ------------------------------------- per-row radix select (rank 819 desc)
__global__ __launch_bounds__(256) void koleo_select(const float* __restrict__ d2,
                                                    int* __restrict__ outIdx) {
  __shared__ uint32_t keys[NN];       // 32 KB (LDS: 320 KB/WGP on CDNA5)
  __shared__ uint32_t hist[256];
  __shared__ uint32_t cnts[256];
  __shared__ uint32_t s_prefix;
  __shared__ int s_rank;

  const int tid = threadIdx.x;
  const int row = blockIdx.x;
  const float* rp = d2 + (size_t)row * NN;

  // float -> order-preserving uint key (ascending); NT load: each d^2 element
  // is consumed exactly once
  for (int j = tid; j < NN; j += 256) {
    uint32_t b = __float_as_uint(__builtin_nontemporal_load(&rp[j]));
    keys[j] = b ^ (((int32_t)b < 0) ? 0xFFFFFFFFu : 0x80000000u);
  }
  if (tid == 0) { s_prefix = 0u; s_rank = RANK_ASC; }
  __syncthreads();

#pragma unroll
  for (int pass = 0; pass < 4; ++pass) {
    const int shift = 24 - 8 * pass;
    hist[tid] = 0;
    __syncthreads();
    const uint32_t pref  = s_prefix;
    const uint32_t pmask = (pass == 0) ? 0u : (0xFFFFFFFFu << (shift + 8));
    for (int j = tid; j < NN; j += 256) {
      uint32_t k = keys[j];
      if ((k & pmask) == pref) atomicAdd(&hist[(k >> shift) & 0xFFu], 1u);
    }
    __syncthreads();
    if (tid == 0) {
      uint32_t rr = (uint32_t)s_rank, cum = 0;
      int b = 0;
      for (; b < 256; ++b) {
        uint32_t h = hist[b];
        if (cum + h > rr) break;
        cum += h;
      }
      if (b == 256) b = 255;  // unreachable guard
      s_rank = (int)(rr - cum);
      s_prefix = pref | ((uint32_t)b << shift);
    }
    __syncthreads();
  }

  const uint32_t key = s_prefix;      // exact selected key
  const int offAsc  = s_rank;         // rank within equal-key group (ascending)

  // count equals per contiguous 32-chunk -> exclusive scan -> pick occurrence
  const int base = tid * (NN / 256);
  uint32_t cEq = 0;
#pragma unroll 8
  for (int j = 0; j < NN / 256; ++j) cEq += (keys[base + j] == key);
  cnts[tid] = cEq;
  __syncthreads();
  if (tid == 0) {
    uint32_t run = 0;
    for (int t = 0; t < 256; ++t) { uint32_t c = cnts[t]; cnts[t] = run; run += c; }
    hist[0] = run;  // total equal count
  }
  __syncthreads();
  // stable descending argsort: within equal group pick (total-1-offAsc)-th
  // occurrence in ascending index order
  const int m = (int)hist[0] - 1 - offAsc;
  const uint32_t myPre = cnts[tid];
  if (m >= (int)myPre && m < (int)(myPre + cEq)) {
    int need = m - (int)myPre;
    for (int j = 0; j < NN / 256; ++j) {
      if (keys[base + j] == key) {
        if (need == 0) { outIdx[row] = base + j; break; }
        --need;
      }
    }
  }
}

// ------------------------------------------------------- loss, deterministic
__global__ __launch_bounds__(256) void koleo_loss_partial(const float* __restrict__ xi,
                                                          const float* __restrict__ xj,
                                                          const int* __restrict__ I,
                                                          float* __restrict__ partials) {
  __shared__ float wsums[8];
  const int wave = threadIdx.x >> 5, lane = threadIdx.x & 31;
  float acc = 0.f;
  for (int row = blockIdx.x * 8 + wave; row < NN; row += gridDim.x * 8) {
    const int idx = I[row];
    const float* a = xi + (size_t)row * DD + lane;
    const float* b = xj + (size_t)idx * DD + lane;
    float s = 0.f;
#pragma unroll
    for (int t = 0; t < DD / 32; ++t) {
      float df = a[t * 32] - b[t * 32] + 1e-8f;  // + EPS before the norm
      s += df * df;
    }
    s = wave_reduce_sum(s);
    if (lane == 0) acc += 1.f / (s + 1.f);
  }
  if (lane == 0) wsums[wave] = acc;
  __syncthreads();
  if (threadIdx.x == 0) {
    float t = 0.f;
#pragma unroll
    for (int w = 0; w < 8; ++w) t += wsums[w];
    partials[blockIdx.x] = t;
  }
}

__global__ void koleo_finalize(const float* __restrict__ partials,
                               float* __restrict__ out) {
  const int lane = threadIdx.x & 31;
  float v = partials[lane] + partials[lane + 32];
  v = wave_reduce_sum(v);
  if (lane == 0) out[0] = v / (float)NN;
}

// ------------------------------------------------------------------- launch
extern "C" void kernel_launch(void* const* d_in, const int* in_sizes, int n_in,
                              void* d_out, int out_size, void* d_ws, size_t ws_size,
                              hipStream_t stream) {
  (void)in_sizes; (void)n_in; (void)out_size; (void)ws_size;
  const float* xi = (const float*)d_in[0];
  const float* xj = (const float*)d_in[1];
  float* ws = (float*)d_ws;
  float* ni = ws;                       // [NN]
  float* nj = ws + NN;                  // [NN]
  int*   I  = (int*)(ws + 2 * NN);      // [NN]
  float* partials = ws + 3 * NN;        // [64]
  float* d2 = ws + 4 * NN;              // [NN*NN] = 256 MiB, 128KB-aligned
  float* out = (float*)d_out;

  koleo_norms<<<(2 * NN) / 8, 256, 0, stream>>>(xi, xj, ni, nj);
  dim3 g(NN / 512, NN / 16);
  koleo_gemm_d2<<<g, 256, 0, stream>>>(xi, xj, ni, nj, d2);
  koleo_select<<<NN, 256, 0, stream>>>(d2, I);
  koleo_loss_partial<<<64, 256, 0, stream>>>(xi, xj, I, partials);
  koleo_finalize<<<1, 32, 0, stream>>>(partials, out);
}